// HetGNN_83313775607884
// MI455X (gfx1250) — compile-verified
//
#include <hip/hip_runtime.h>
#include <math.h>

// ---------------------------------------------------------------------------
// HetGNN on MI455X (gfx1250):
//   - relation messages rewritten as scatter-then-GEMM (linearity of segment_sum)
//   - fixed relations (lane/sens/inj) aggregated on raw x (16/8/12 floats/edge)
//   - per layer: one 640K-edge H=128 float scatter + ONE fused K=512 WMMA GEMM
//     out = ELU([h | mean_sp | m_flow | m_inj] @ [Wself;Wsp;Wfl;Winc] + b)
// ---------------------------------------------------------------------------

typedef __attribute__((ext_vector_type(2))) float v2f;
typedef __attribute__((ext_vector_type(8))) float v8f;

#define HDIM 128
#define BM 64
#define BK 32

// ---- input projection for h_int: [n,32] @ [32,128] + b -------------------
__global__ __launch_bounds__(256) void proj_int_kernel(
    const float* __restrict__ x, const float* __restrict__ W,
    const float* __restrict__ b, float* __restrict__ out, int n) {
  int t = blockIdx.x * blockDim.x + threadIdx.x;
  int node = t >> 7, col = t & 127;
  if (node >= n) return;
  const float* xr = x + (size_t)node * 32;
  float acc = b[col];
#pragma unroll
  for (int k = 0; k < 32; ++k) acc = fmaf(xr[k], W[k * HDIM + col], acc);
  out[(size_t)node * HDIM + col] = acc;
}

// ---- per-target edge counts ----------------------------------------------
__global__ __launch_bounds__(256) void count_kernel(
    const int* __restrict__ dst, float* __restrict__ cnt, int e) {
  int t = blockIdx.x * blockDim.x + threadIdx.x;
  if (t < e) atomicAdd(&cnt[dst[t]], 1.0f);
}

// ---- scatter-add of raw small-dim inputs (K = 16 / 8 / 12) ---------------
__global__ __launch_bounds__(256) void scatter_x_kernel(
    const int* __restrict__ edges, const float* __restrict__ x,
    float* __restrict__ agg, int e, int K) {
  int t = blockIdx.x * blockDim.x + threadIdx.x;
  if (t >= e) return;
  int s = edges[t], d = edges[e + t];
  const float* xr = x + (size_t)s * K;
  float* ar = agg + (size_t)d * K;
  for (int k = 0; k < K; ++k) atomicAdd(&ar[k], xr[k]);
}

// ---- scatter-add of full H=128 rows of h_int (per layer) -----------------
__global__ __launch_bounds__(256) void scatter_h_kernel(
    const int* __restrict__ edges, const float* __restrict__ h,
    float* __restrict__ agg, int e) {
  int t = blockIdx.x * blockDim.x + threadIdx.x;
  int eg = t >> 5;               // one edge handled by 32 threads
  int c4 = (t & 31) << 2;        // 4 columns each
  if (eg >= e) return;
  int s = edges[eg], d = edges[e + eg];
  float4 v = *(const float4*)(h + (size_t)s * HDIM + c4);
  float* ap = agg + (size_t)d * HDIM + c4;
  atomicAdd(ap + 0, v.x);
  atomicAdd(ap + 1, v.y);
  atomicAdd(ap + 2, v.z);
  atomicAdd(ap + 3, v.w);
}

// ---- m_flow = mean_lane + mean_sens via affine trick (computed once) -----
__global__ __launch_bounds__(256) void mflow_kernel(
    const float* __restrict__ aggL, const float* __restrict__ cntL,
    const float* __restrict__ WL, const float* __restrict__ bL,
    const float* __restrict__ aggS, const float* __restrict__ cntS,
    const float* __restrict__ WS, const float* __restrict__ bS,
    float* __restrict__ out, int n) {
  int t = blockIdx.x * blockDim.x + threadIdx.x;
  int node = t >> 7, col = t & 127;
  if (node >= n) return;
  float cl = cntL[node], cs = cntS[node];
  float a = 0.f;
  const float* al = aggL + (size_t)node * 16;
#pragma unroll
  for (int k = 0; k < 16; ++k) a = fmaf(al[k], WL[k * HDIM + col], a);
  a *= 1.0f / fmaxf(cl, 1.0f);
  if (cl > 0.f) a += bL[col];
  float b2 = 0.f;
  const float* as = aggS + (size_t)node * 8;
#pragma unroll
  for (int k = 0; k < 8; ++k) b2 = fmaf(as[k], WS[k * HDIM + col], b2);
  b2 *= 1.0f / fmaxf(cs, 1.0f);
  if (cs > 0.f) b2 += bS[col];
  out[(size_t)node * HDIM + col] = a + b2;
}

// ---- m_inj = mean_incident via affine trick (computed once) --------------
__global__ __launch_bounds__(256) void minj_kernel(
    const float* __restrict__ aggI, const float* __restrict__ cntI,
    const float* __restrict__ WI, const float* __restrict__ bI,
    float* __restrict__ out, int n) {
  int t = blockIdx.x * blockDim.x + threadIdx.x;
  int node = t >> 7, col = t & 127;
  if (node >= n) return;
  float ci = cntI[node];
  float a = 0.f;
  const float* ai = aggI + (size_t)node * 12;
#pragma unroll
  for (int k = 0; k < 12; ++k) a = fmaf(ai[k], WI[k * HDIM + col], a);
  a *= 1.0f / fmaxf(ci, 1.0f);
  if (ci > 0.f) a += bI[col];
  out[(size_t)node * HDIM + col] = a;
}

// ---- fused layer GEMM: [h|mean_sp|m_flow|m_inj](64x512) @ Wcat(512x128) --
// 8 waves/block, each wave: one 16-col N strip x four 16-row M tiles.
// v_wmma_f32_16x16x4_f32, K staged through LDS in 32-slabs.
__global__ __launch_bounds__(256) void fused_layer_gemm(
    const float* __restrict__ hcur, const float* __restrict__ aggsp,
    const float* __restrict__ cnt_sp, const float* __restrict__ mflow,
    const float* __restrict__ minj,
    const float* __restrict__ Wself, const float* __restrict__ Wsp,
    const float* __restrict__ Wfl, const float* __restrict__ Winc,
    const float* __restrict__ bias, float* __restrict__ out, int n) {
  __shared__ float As[BM][BK + 1];        // 64 x 33
  __shared__ float Bs[BK][HDIM + 4];      // 32 x 132
  __shared__ float invs[BM];

  const int tid = threadIdx.x;
  const int wave = tid >> 5;
  const int lane = tid & 31;
  const int half = lane >> 4;   // K sub-pair select
  const int nl = lane & 15;     // row (A) / col (B,D) within tile
  const int row0 = blockIdx.x * BM;

  if (tid < BM) {
    int r = row0 + tid;
    float c = (r < n) ? cnt_sp[r] : 1.0f;
    invs[tid] = 1.0f / fmaxf(c, 1.0f);
  }
  __syncthreads();

  v8f acc[4];
#pragma unroll
  for (int m = 0; m < 4; ++m)
#pragma unroll
    for (int j = 0; j < 8; ++j) acc[m][j] = 0.0f;

  // K = 512 total: segment 0=h@Wself, 1=mean_sp@Wsp, 2=m_flow@Wfl, 3=m_inj@Winc
#pragma unroll 1
  for (int kb = 0; kb < 16; ++kb) {
    const int k0 = kb * BK;
    const int seg = k0 >> 7;
    const int ks = k0 & 127;
    const float* Asrc;
    const float* Bsrc;
    bool scale = false;
    if (seg == 0)      { Asrc = hcur;  Bsrc = Wself; }
    else if (seg == 1) { Asrc = aggsp; Bsrc = Wsp; scale = true; }
    else if (seg == 2) { Asrc = mflow; Bsrc = Wfl; }
    else               { Asrc = minj;  Bsrc = Winc; }

    // stage A: 64x32 floats (512 float4 slots, 2 per thread)
#pragma unroll
    for (int i = 0; i < 2; ++i) {
      int slot = tid + i * 256;
      int r = slot >> 3;
      int c4 = (slot & 7) << 2;
      int gr = row0 + r;
      float4 v = make_float4(0.f, 0.f, 0.f, 0.f);
      if (gr < n) v = *(const float4*)(Asrc + (size_t)gr * HDIM + ks + c4);
      if (scale) {
        float s = invs[r];
        v.x *= s; v.y *= s; v.z *= s; v.w *= s;
      }
      As[r][c4 + 0] = v.x; As[r][c4 + 1] = v.y;
      As[r][c4 + 2] = v.z; As[r][c4 + 3] = v.w;
    }
    // stage B: 32x128 floats (1024 float4 slots, 4 per thread)
#pragma unroll
    for (int i = 0; i < 4; ++i) {
      int slot = tid + i * 256;
      int kr = slot >> 5;
      int c4 = (slot & 31) << 2;
      float4 v = *(const float4*)(Bsrc + (size_t)(ks + kr) * HDIM + c4);
      Bs[kr][c4 + 0] = v.x; Bs[kr][c4 + 1] = v.y;
      Bs[kr][c4 + 2] = v.z; Bs[kr][c4 + 3] = v.w;
    }
    __syncthreads();

#pragma unroll
    for (int kk = 0; kk < BK; kk += 4) {
      const int kA = kk + half * 2;   // lanes 0-15: K{0,1}; lanes 16-31: K{2,3}
      v2f b;
      b.x = Bs[kA + 0][wave * 16 + nl];
      b.y = Bs[kA + 1][wave * 16 + nl];
#pragma unroll
      for (int m = 0; m < 4; ++m) {
        v2f a;
        a.x = As[m * 16 + nl][kA + 0];
        a.y = As[m * 16 + nl][kA + 1];
        acc[m] = __builtin_amdgcn_wmma_f32_16x16x4_f32(
            false, a, false, b, (short)0, acc[m], false, false);
      }
    }
    __syncthreads();
  }

  // epilogue: + b_self, ELU, store. D layout: VGPR v -> row (half*8+v), col nl.
  const int col = wave * 16 + nl;
  const float bc = bias[col];
#pragma unroll
  for (int m = 0; m < 4; ++m) {
#pragma unroll
    for (int v = 0; v < 8; ++v) {
      int r = row0 + m * 16 + half * 8 + v;
      if (r < n) {
        float x = acc[m][v] + bc;
        out[(size_t)r * HDIM + col] = (x > 0.f) ? x : expm1f(x);
      }
    }
  }
}

// ---------------------------------------------------------------------------
extern "C" void kernel_launch(void* const* d_in, const int* in_sizes, int n_in,
                              void* d_out, int out_size, void* d_ws, size_t ws_size,
                              hipStream_t stream) {
  const float* x_int  = (const float*)d_in[0];
  const float* x_lane = (const float*)d_in[1];
  const float* x_sens = (const float*)d_in[2];
  const float* x_inj  = (const float*)d_in[3];
  const float* Wp_int  = (const float*)d_in[4];
  const float* bp_int  = (const float*)d_in[5];
  const float* Wp_lane = (const float*)d_in[6];
  const float* bp_lane = (const float*)d_in[7];
  const float* Wp_sens = (const float*)d_in[8];
  const float* bp_sens = (const float*)d_in[9];
  const float* Wp_inj  = (const float*)d_in[10];
  const float* bp_inj  = (const float*)d_in[11];
  const float* W_self     = (const float*)d_in[12];
  const float* b_self     = (const float*)d_in[13];
  const float* W_spatial  = (const float*)d_in[14];
  const float* W_flow     = (const float*)d_in[15];
  const float* W_incident = (const float*)d_in[16];
  const int* sp_e = (const int*)d_in[17];
  const int* fl_e = (const int*)d_in[18];
  const int* fs_e = (const int*)d_in[19];
  const int* in_e = (const int*)d_in[20];

  const int n    = in_sizes[0] / 32;
  const int E_sp = in_sizes[17] / 2;
  const int E_fl = in_sizes[18] / 2;
  const int E_fs = in_sizes[19] / 2;
  const int E_in = in_sizes[20] / 2;

  // workspace carve-out
  size_t off = 0;
  auto carve = [&](size_t bytes) -> float* {
    float* p = (float*)((char*)d_ws + off);
    off += (bytes + 255) & ~(size_t)255;
    return p;
  };
  const size_t nh = (size_t)n * HDIM * sizeof(float);
  float* h_a    = carve(nh);
  float* aggsp  = carve(nh);
  float* mflow  = carve(nh);
  float* minj_b = carve(nh);
  float* aggL   = carve((size_t)n * 16 * sizeof(float));
  float* aggS   = carve((size_t)n * 8 * sizeof(float));
  float* aggI   = carve((size_t)n * 12 * sizeof(float));
  float* cnt_sp = carve((size_t)n * sizeof(float));
  float* cnt_l  = carve((size_t)n * sizeof(float));
  float* cnt_s  = carve((size_t)n * sizeof(float));
  float* cnt_i  = carve((size_t)n * sizeof(float));

  hipMemsetAsync(aggL,   0, (size_t)n * 16 * sizeof(float), stream);
  hipMemsetAsync(aggS,   0, (size_t)n * 8  * sizeof(float), stream);
  hipMemsetAsync(aggI,   0, (size_t)n * 12 * sizeof(float), stream);
  hipMemsetAsync(cnt_sp, 0, (size_t)n * sizeof(float), stream);
  hipMemsetAsync(cnt_l,  0, (size_t)n * sizeof(float), stream);
  hipMemsetAsync(cnt_s,  0, (size_t)n * sizeof(float), stream);
  hipMemsetAsync(cnt_i,  0, (size_t)n * sizeof(float), stream);

  // h_int projection
  {
    int thr = n * HDIM;
    proj_int_kernel<<<(thr + 255) / 256, 256, 0, stream>>>(x_int, Wp_int, bp_int, h_a, n);
  }

  // per-target counts for all relations
  count_kernel<<<(E_sp + 255) / 256, 256, 0, stream>>>(sp_e + E_sp, cnt_sp, E_sp);
  count_kernel<<<(E_fl + 255) / 256, 256, 0, stream>>>(fl_e + E_fl, cnt_l,  E_fl);
  count_kernel<<<(E_fs + 255) / 256, 256, 0, stream>>>(fs_e + E_fs, cnt_s,  E_fs);
  count_kernel<<<(E_in + 255) / 256, 256, 0, stream>>>(in_e + E_in, cnt_i,  E_in);

  // fixed-relation aggregates on raw x (cheap: 16/8/12 floats per edge)
  scatter_x_kernel<<<(E_fl + 255) / 256, 256, 0, stream>>>(fl_e, x_lane, aggL, E_fl, 16);
  scatter_x_kernel<<<(E_fs + 255) / 256, 256, 0, stream>>>(fs_e, x_sens, aggS, E_fs, 8);
  scatter_x_kernel<<<(E_in + 255) / 256, 256, 0, stream>>>(in_e, x_inj,  aggI, E_in, 12);

  // fixed means projected to H (once; reused by all 3 layers)
  {
    int thr = n * HDIM;
    mflow_kernel<<<(thr + 255) / 256, 256, 0, stream>>>(
        aggL, cnt_l, Wp_lane, bp_lane, aggS, cnt_s, Wp_sens, bp_sens, mflow, n);
    minj_kernel<<<(thr + 255) / 256, 256, 0, stream>>>(
        aggI, cnt_i, Wp_inj, bp_inj, minj_b, n);
  }

  // layers: ping-pong h_a <-> d_out (L=3 ends in d_out)
  float* hcur = h_a;
  float* hout = (float*)d_out;
  const size_t HH = (size_t)HDIM * HDIM;
  for (int l = 0; l < 3; ++l) {
    hipMemsetAsync(aggsp, 0, nh, stream);
    int thr = E_sp * 32;
    scatter_h_kernel<<<(thr + 255) / 256, 256, 0, stream>>>(sp_e, hcur, aggsp, E_sp);
    fused_layer_gemm<<<(n + BM - 1) / BM, 256, 0, stream>>>(
        hcur, aggsp, cnt_sp, mflow, minj_b,
        W_self + l * HH, W_spatial + l * HH, W_flow + l * HH, W_incident + l * HH,
        b_self + (size_t)l * HDIM, hout, n);
    float* t = hcur; hcur = hout; hout = t;
  }
}